// GraphSageLayer_2001454759950
// MI455X (gfx1250) — compile-verified
//
#include <hip/hip_runtime.h>

#define NN 40000
#define EE 640000
#define DD 128

typedef __attribute__((ext_vector_type(2))) float v2f;
typedef __attribute__((ext_vector_type(8))) float v8f;

// ---------------- zero-init workspace (h_sum, deg, colsum, colsq) -----------
__global__ __launch_bounds__(256) void init_zero(float* __restrict__ p, long n) {
    long i = (long)blockIdx.x * blockDim.x + threadIdx.x;
    long stride = (long)gridDim.x * blockDim.x;
    for (; i < n; i += stride) p[i] = 0.0f;
}

// ---------------- edge scatter: one wave32 per edge -------------------------
// lane k loads float4 of h[src[e]] and atomically adds to h_sum[dst[e]].
__global__ __launch_bounds__(256) void edge_scatter(const float* __restrict__ h,
                                                    const int* __restrict__ src,
                                                    const int* __restrict__ dst,
                                                    float* __restrict__ hsum,
                                                    float* __restrict__ deg) {
    int t = blockIdx.x * 256 + threadIdx.x;
    int e = t >> 5;
    int lane = t & 31;
    if (e >= EE) return;
    int s = src[e];
    int d = dst[e];
    float4 v = ((const float4*)(h + (size_t)s * DD))[lane];
    float* dp = hsum + (size_t)d * DD + lane * 4;
    atomicAdd(dp + 0, v.x);
    atomicAdd(dp + 1, v.y);
    atomicAdd(dp + 2, v.z);
    atomicAdd(dp + 3, v.w);
    if (lane == 0) atomicAdd(deg + d, 1.0f);
}

// ---------------- fused GEMM: out = relu(h@Ws + (hsum/deg)@Wn + b) ----------
// One wave computes a 16x16 tile via V_WMMA_F32_16X16X4_F32 (K=256 fused).
// Epilogue stores relu output and accumulates per-column sum / sum-of-squares.
__global__ __launch_bounds__(256) void sage_gemm(const float* __restrict__ h,
                                                 const float* __restrict__ hsum,
                                                 const float* __restrict__ deg,
                                                 const float* __restrict__ Wself,
                                                 const float* __restrict__ Wneigh,
                                                 const float* __restrict__ bias,
                                                 float* __restrict__ tmp,
                                                 float* __restrict__ colsum,
                                                 float* __restrict__ colsq) {
    const int lane = threadIdx.x & 31;
    const int wave = threadIdx.x >> 5;   // 8 waves -> 8 column tiles (128 cols)
    const int row0 = blockIdx.x << 4;    // 2500 blocks * 16 rows = 40000 exactly
    const int col0 = wave << 4;
    const int m    = lane & 15;          // A row / B,C col within tile
    const int half = lane >> 4;          // K sub-pair select (ISA A/B layout)

    const float* arow = h    + (size_t)(row0 + m) * DD;
    const float* srow = hsum + (size_t)(row0 + m) * DD;
    const float  dinv = 1.0f / fmaxf(deg[row0 + m], 1.0f);

    v8f acc = {};
#pragma unroll 8
    for (int k = 0; k < DD; k += 4) {
        const int ka = k + half * 2;
        v2f a = *(const v2f*)(arow + ka);
        v2f b;
        b.x = Wself[(size_t)ka * DD + col0 + m];
        b.y = Wself[(size_t)(ka + 1) * DD + col0 + m];
        acc = __builtin_amdgcn_wmma_f32_16x16x4_f32(false, a, false, b,
                                                    (short)0, acc, false, false);
    }
#pragma unroll 8
    for (int k = 0; k < DD; k += 4) {
        const int ka = k + half * 2;
        v2f a = *(const v2f*)(srow + ka);
        a.x *= dinv;
        a.y *= dinv;
        v2f b;
        b.x = Wneigh[(size_t)ka * DD + col0 + m];
        b.y = Wneigh[(size_t)(ka + 1) * DD + col0 + m];
        acc = __builtin_amdgcn_wmma_f32_16x16x4_f32(false, a, false, b,
                                                    (short)0, acc, false, false);
    }

    const float bn = bias[col0 + m];
    float ps = 0.0f, pq = 0.0f;
#pragma unroll
    for (int v = 0; v < 8; ++v) {
        float val = fmaxf(acc[v] + bn, 0.0f);          // D layout: M = v + 8*half
        tmp[(size_t)(row0 + half * 8 + v) * DD + col0 + m] = val;
        ps += val;
        pq += val * val;
    }
    atomicAdd(colsum + col0 + m, ps);
    atomicAdd(colsq  + col0 + m, pq);
}

// ---------------- BN stats -> fused affine (scale/shift) --------------------
__global__ void bn_stats(const float* __restrict__ colsum,
                         const float* __restrict__ colsq,
                         const float* __restrict__ gamma,
                         const float* __restrict__ beta,
                         float* __restrict__ scale,
                         float* __restrict__ shift) {
    int d = threadIdx.x;
    const float inv_n = 1.0f / (float)NN;
    float mu  = colsum[d] * inv_n;
    float var = colsq[d] * inv_n - mu * mu;           // biased variance
    float rs  = rsqrtf(fmaxf(var, 0.0f) + 1e-5f);
    float sc  = gamma[d] * rs;
    scale[d] = sc;
    shift[d] = beta[d] - mu * sc;
}

// ---------------- residual + BN apply: out = h + tmp*scale + shift ----------
__global__ __launch_bounds__(256) void bn_apply(const float* __restrict__ h,
                                                const float* __restrict__ tmp,
                                                const float* __restrict__ scale,
                                                const float* __restrict__ shift,
                                                float* __restrict__ out) {
    size_t i = ((size_t)blockIdx.x * 256 + threadIdx.x) * 4;
    int d = (int)(i & (DD - 1));
    float4 hv = *(const float4*)(h + i);
    float4 tv = *(const float4*)(tmp + i);
    float4 o;
    o.x = hv.x + tv.x * scale[d]     + shift[d];
    o.y = hv.y + tv.y * scale[d + 1] + shift[d + 1];
    o.z = hv.z + tv.z * scale[d + 2] + shift[d + 2];
    o.w = hv.w + tv.w * scale[d + 3] + shift[d + 3];
    *(float4*)(out + i) = o;
}

extern "C" void kernel_launch(void* const* d_in, const int* in_sizes, int n_in,
                              void* d_out, int out_size, void* d_ws, size_t ws_size,
                              hipStream_t stream) {
    const float* h      = (const float*)d_in[0];
    const int*   src    = (const int*)d_in[1];
    const int*   dst    = (const int*)d_in[2];
    const float* Wself  = (const float*)d_in[3];
    const float* Wneigh = (const float*)d_in[4];
    const float* bias   = (const float*)d_in[5];
    const float* gamma  = (const float*)d_in[6];
    const float* beta   = (const float*)d_in[7];
    float* out = (float*)d_out;

    float* ws     = (float*)d_ws;
    float* hsum   = ws;                            // N*D
    float* deg    = hsum + (size_t)NN * DD;        // N
    float* colsum = deg + NN;                      // D
    float* colsq  = colsum + DD;                   // D
    float* scale  = colsq + DD;                    // D
    float* shift  = scale + DD;                    // D
    float* tmp    = shift + DD;                    // N*D  (relu output)

    long zcount = (long)NN * DD + NN + 2 * DD;     // hsum | deg | colsum | colsq
    init_zero<<<2048, 256, 0, stream>>>(ws, zcount);

    edge_scatter<<<(EE * 32) / 256, 256, 0, stream>>>(h, src, dst, hsum, deg);

    sage_gemm<<<NN / 16, 256, 0, stream>>>(h, hsum, deg, Wself, Wneigh, bias,
                                           tmp, colsum, colsq);

    bn_stats<<<1, DD, 0, stream>>>(colsum, colsq, gamma, beta, scale, shift);

    bn_apply<<<(NN * DD / 4) / 256, 256, 0, stream>>>(h, tmp, scale, shift, out);
}